// ResidualBlock_12180527251932
// MI455X (gfx1250) — compile-verified
//
#include <hip/hip_runtime.h>

#define NN 50000
#define DD 256
#define NE 300000
#define LDST 260   // padded LDS row stride (floats): 16B-aligned rows, banks spread 4 apart

typedef __attribute__((ext_vector_type(2))) float v2f;
typedef __attribute__((ext_vector_type(8))) float v8f;

// ---------------------------------------------------------------- zero int region (hist + cursors)
__global__ void zero_kernel(int* __restrict__ p, int n) {
    int i = blockIdx.x * blockDim.x + threadIdx.x;
    if (i < n) p[i] = 0;
}

// ---------------------------------------------------------------- degree histogram
__global__ void hist_kernel(const int* __restrict__ dst, int* __restrict__ hist) {
    int e = blockIdx.x * blockDim.x + threadIdx.x;
    if (e < NE) atomicAdd(&hist[dst[e]], 1);
}

// ---------------------------------------------------------------- exclusive scan over NN bins (single block)
__global__ __launch_bounds__(1024)
void scan_kernel(const int* __restrict__ hist, int* __restrict__ offs) {
    __shared__ int buf[1024];
    __shared__ int carry;
    const int tid = threadIdx.x;
    if (tid == 0) carry = 0;
    __syncthreads();
    for (int base = 0; base < NN; base += 1024) {
        int i = base + tid;
        int v = (i < NN) ? hist[i] : 0;
        buf[tid] = v;
        __syncthreads();
        for (int off = 1; off < 1024; off <<= 1) {
            int t = (tid >= off) ? buf[tid - off] : 0;
            __syncthreads();
            buf[tid] += t;
            __syncthreads();
        }
        int incl = buf[tid];
        if (i < NN) offs[i] = carry + incl - v;   // exclusive
        __syncthreads();                          // everyone read carry
        if (tid == 1023) carry += incl;           // chunk total
        __syncthreads();
    }
    if (tid == 0) offs[NN] = carry;               // == NE
}

// ---------------------------------------------------------------- bucket src ids by dst
__global__ void bucket_kernel(const int* __restrict__ src, const int* __restrict__ dst,
                              const int* __restrict__ offs, int* __restrict__ cursor,
                              int* __restrict__ ssrc) {
    int e = blockIdx.x * blockDim.x + threadIdx.x;
    if (e >= NE) return;
    int d = dst[e];
    int pos = offs[d] + atomicAdd(&cursor[d], 1);
    ssrc[pos] = src[e];
}

// ---------------------------------------------------------------- atomic-free mean aggregation: 1 wave / node
__global__ __launch_bounds__(256)
void aggregate_kernel(const float* __restrict__ x, const int* __restrict__ ssrc,
                      const int* __restrict__ offs, float* __restrict__ mean) {
    const int wave = threadIdx.x >> 5;
    const int lane = threadIdx.x & 31;
    const int n = blockIdx.x * 8 + wave;          // 50000 = 6250*8, always valid
    const int beg = offs[n], end = offs[n + 1];
    const int c = lane * 8;                        // 32 lanes x 8 floats = 256
    float4 a0 = make_float4(0.f, 0.f, 0.f, 0.f);
    float4 a1 = make_float4(0.f, 0.f, 0.f, 0.f);
    for (int j = beg; j < end; ++j) {
        const float* row = x + (size_t)ssrc[j] * DD + c;
        float4 v0 = *(const float4*)(row);
        float4 v1 = *(const float4*)(row + 4);
        a0.x += v0.x; a0.y += v0.y; a0.z += v0.z; a0.w += v0.w;
        a1.x += v1.x; a1.y += v1.y; a1.z += v1.z; a1.w += v1.w;
    }
    const float inv = 1.0f / fmaxf((float)(end - beg), 1.0f);
    a0.x *= inv; a0.y *= inv; a0.z *= inv; a0.w *= inv;
    a1.x *= inv; a1.y *= inv; a1.z *= inv; a1.w *= inv;
    float* o = mean + (size_t)n * DD + c;
    *(float4*)(o)     = a0;
    *(float4*)(o + 4) = a1;
}

// ---------------------------------------------------------------- h = relu(mean@W_l^T + b_l + x@W_r^T)
__global__ __launch_bounds__(256)
void sage_gemm_kernel(const float* __restrict__ mean,
                      const float* __restrict__ x,
                      const float* __restrict__ W_l,
                      const float* __restrict__ b_l,
                      const float* __restrict__ W_r,
                      float* __restrict__ h) {
    __shared__ float sMean[16 * LDST];
    __shared__ float sX[16 * LDST];
    const int tid  = threadIdx.x;
    const int row0 = blockIdx.x * 16;

    {   // stage 16x256 strips of mean and x into LDS
        int r  = tid >> 4;
        int cb = (tid & 15) << 4;
        const float* gm = mean + (size_t)(row0 + r) * DD + cb;
        const float* gx = x    + (size_t)(row0 + r) * DD + cb;
        float* lm = sMean + r * LDST + cb;
        float* lx = sX    + r * LDST + cb;
#pragma unroll
        for (int i = 0; i < 16; i += 4) {
            *(float4*)(lm + i) = *(const float4*)(gm + i);
            *(float4*)(lx + i) = *(const float4*)(gx + i);
        }
    }
    __syncthreads();

    const int lane = tid & 31;
    const int wave = tid >> 5;
    const int m    = lane & 15;         // row-in-tile (A) / col-in-tile (B,C)
    const int koff = (lane >> 4) << 1;  // 0 or 2

#pragma unroll
    for (int t = 0; t < 2; ++t) {
        const int col0 = (wave * 2 + t) * 16;
        const float* wl = W_l + (size_t)(col0 + m) * DD + koff;
        const float* wr = W_r + (size_t)(col0 + m) * DD + koff;
        const float* am = sMean + m * LDST + koff;
        const float* ax = sX    + m * LDST + koff;
        v8f accL = {0.f,0.f,0.f,0.f,0.f,0.f,0.f,0.f};
        v8f accR = {0.f,0.f,0.f,0.f,0.f,0.f,0.f,0.f};
#pragma unroll 8
        for (int k = 0; k < DD; k += 4) {
            v2f a0 = *(const v2f*)(am + k);
            v2f b0 = *(const v2f*)(wl + k);
            accL = __builtin_amdgcn_wmma_f32_16x16x4_f32(false, a0, false, b0,
                                                         (short)0, accL, false, false);
            v2f a1 = *(const v2f*)(ax + k);
            v2f b1 = *(const v2f*)(wr + k);
            accR = __builtin_amdgcn_wmma_f32_16x16x4_f32(false, a1, false, b1,
                                                         (short)0, accR, false, false);
        }
        const float bias = b_l[col0 + m];
        const int rbase  = row0 + ((lane >> 4) << 3);
        float* out = h + (size_t)rbase * DD + col0 + m;
#pragma unroll
        for (int v = 0; v < 8; ++v)
            out[(size_t)v * DD] = fmaxf(accL[v] + accR[v] + bias, 0.0f);
    }
}

// ---------------------------------------------------------------- out = relu(h@W_ln^T + x), in place over h
__global__ __launch_bounds__(256)
void ln_res_kernel(const float* __restrict__ x,
                   const float* __restrict__ W_ln,
                   float* __restrict__ out) {
    __shared__ float sH[16 * LDST];
    const int tid  = threadIdx.x;
    const int row0 = blockIdx.x * 16;

    {   // stage h strip from d_out (so in-place overwrite below is safe)
        int r  = tid >> 4;
        int cb = (tid & 15) << 4;
        const float* gh = out + (size_t)(row0 + r) * DD + cb;
        float* lh = sH + r * LDST + cb;
#pragma unroll
        for (int i = 0; i < 16; i += 4)
            *(float4*)(lh + i) = *(const float4*)(gh + i);
    }
    __syncthreads();

    const int lane = tid & 31;
    const int wave = tid >> 5;
    const int m    = lane & 15;
    const int koff = (lane >> 4) << 1;

#pragma unroll
    for (int t = 0; t < 2; ++t) {
        const int col0 = (wave * 2 + t) * 16;
        const float* wn = W_ln + (size_t)(col0 + m) * DD + koff;
        const float* ah = sH + m * LDST + koff;
        v8f acc0 = {0.f,0.f,0.f,0.f,0.f,0.f,0.f,0.f};
        v8f acc1 = {0.f,0.f,0.f,0.f,0.f,0.f,0.f,0.f};
#pragma unroll 8
        for (int k = 0; k < DD; k += 8) {
            v2f a0 = *(const v2f*)(ah + k);
            v2f b0 = *(const v2f*)(wn + k);
            acc0 = __builtin_amdgcn_wmma_f32_16x16x4_f32(false, a0, false, b0,
                                                         (short)0, acc0, false, false);
            v2f a1 = *(const v2f*)(ah + k + 4);
            v2f b1 = *(const v2f*)(wn + k + 4);
            acc1 = __builtin_amdgcn_wmma_f32_16x16x4_f32(false, a1, false, b1,
                                                         (short)0, acc1, false, false);
        }
        const int rbase = row0 + ((lane >> 4) << 3);
        const float* xr = x   + (size_t)rbase * DD + col0 + m;
        float*       o  = out + (size_t)rbase * DD + col0 + m;
#pragma unroll
        for (int v = 0; v < 8; ++v)
            o[(size_t)v * DD] = fmaxf(acc0[v] + acc1[v] + xr[(size_t)v * DD], 0.0f);
    }
}

// ---------------------------------------------------------------- launcher
extern "C" void kernel_launch(void* const* d_in, const int* in_sizes, int n_in,
                              void* d_out, int out_size, void* d_ws, size_t ws_size,
                              hipStream_t stream) {
    const float* x    = (const float*)d_in[0];
    const int*   ei   = (const int*)d_in[1];   // [2, NE] int32 (JAX default demotes int64)
    const float* W_l  = (const float*)d_in[2];
    const float* b_l  = (const float*)d_in[3];
    const float* W_r  = (const float*)d_in[4];
    const float* W_ln = (const float*)d_in[5];
    float* out = (float*)d_out;

    // workspace layout
    float* mean   = (float*)d_ws;                  // NN*DD floats (51.2 MB)
    int*   hist   = (int*)d_ws + (size_t)NN * DD;  // NN
    int*   cursor = hist + NN;                     // NN
    int*   offs   = cursor + NN;                   // NN+1
    int*   ssrc   = offs + NN + 64;                // NE (pad for alignment)

    const int* src = ei;
    const int* dst = ei + NE;

    // 1) zero hist + cursor (2*NN ints)
    zero_kernel<<<(2 * NN + 255) / 256, 256, 0, stream>>>(hist, 2 * NN);

    // 2) degree histogram
    hist_kernel<<<(NE + 255) / 256, 256, 0, stream>>>(dst, hist);

    // 3) exclusive prefix sum -> offs
    scan_kernel<<<1, 1024, 0, stream>>>(hist, offs);

    // 4) bucket src ids by destination
    bucket_kernel<<<(NE + 255) / 256, 256, 0, stream>>>(src, dst, offs, cursor, ssrc);

    // 5) atomic-free mean aggregation (1 wave per node)
    aggregate_kernel<<<NN / 8, 256, 0, stream>>>(x, ssrc, offs, mean);

    // 6) h = relu(mean@W_l^T + b_l + x@W_r^T) -> d_out
    sage_gemm_kernel<<<NN / 16, 256, 0, stream>>>(mean, x, W_l, b_l, W_r, out);

    // 7) out = relu(h@W_ln^T + x) (in place)
    ln_res_kernel<<<NN / 16, 256, 0, stream>>>(x, W_ln, out);
}